// BSplineLayer_24163486008134
// MI455X (gfx1250) — compile-verified
//
#include <hip/hip_runtime.h>
#include <hip/hip_bf16.h>

typedef __attribute__((ext_vector_type(2))) float v2f;
typedef __attribute__((ext_vector_type(8))) float v8f;

#define Z1 (-0.26794919243112270647f)   // -2 + sqrt(3)

// ---------------------------------------------------------------------------
// Stage 1: per-(batch, channel) cyclic cubic-spline control point solve.
// One thread owns one length-64 series, fully register-resident (unrolled).
// ---------------------------------------------------------------------------
__global__ __launch_bounds__(256) void solve_ctrl_pts(
    const float* __restrict__ in, float* __restrict__ Q, int BC) {
  int t = blockIdx.x * blockDim.x + threadIdx.x;
  if (t >= BC) return;
  int b = t >> 1;
  int c = t & 1;
  const float* rp = in + (size_t)b * 128 + c;   // [B,64,2] layout
  float*       qp = Q  + (size_t)b * 128 + c;

  float a[64];
  float p = 1.f, s0 = 0.f;
#pragma unroll
  for (int i = 0; i < 64; ++i) {
    a[i] = rp[2 * i] * 255.f;
    s0 += a[i] * p;
    p *= Z1;
  }
  float denom = 1.f - p;           // 1 - z1^64
  float q = s0 / denom;            // qt0
  a[0] = q;
#pragma unroll
  for (int i = 1; i < 64; ++i) { q = Z1 * q + a[i]; a[i] = q; }
  p = 1.f;
  float s1 = 0.f;
#pragma unroll
  for (int i = 0; i < 64; ++i) { s1 += a[i] * p; p *= Z1; }
  float q0    = -(6.f * Z1 / denom) * s1;
  float qlast = Z1 * q0 - 6.f * Z1 * a[63];
  q = qlast;
#pragma unroll
  for (int i = 62; i >= 1; --i) { q = Z1 * q - 6.f * Z1 * a[i]; a[i] = q; }
  a[0] = q0;
  a[63] = qlast;
  const float inv255 = 1.f / 255.f;
#pragma unroll
  for (int i = 0; i < 64; ++i) qp[2 * i] = a[i] * inv255;
}

// ---------------------------------------------------------------------------
// Stage 2: closed cubic B-spline evaluation via V_WMMA_F32_16X16X4_F32.
// 8 waves per block = 8 tiles = 128 rows = 2560 contiguous output floats.
// D tiles are staged to LDS (64-float padded blocks per (b,seg) pair, so all
// stores are unconditional ds_store_b64), then streamed out as b128 stores.
// ---------------------------------------------------------------------------
__device__ __forceinline__ void basis4(float t, float& b0, float& b1,
                                       float& b2, float& b3) {
  float t2 = t * t, t3 = t2 * t;
  const float k6 = 1.f / 6.f;
  b0 = (-t3 + 3.f * t2 - 3.f * t + 1.f) * k6;
  b1 = (3.f * t3 - 6.f * t2 + 4.f) * k6;
  b2 = (-3.f * t3 + 3.f * t2 + 3.f * t + 1.f) * k6;
  b3 = t3 * k6;
}

__global__ __launch_bounds__(256) void bspline_wmma(
    const float* __restrict__ Q, float* __restrict__ out) {
  __shared__ float lds[64 * 64];               // 64 pairs x 64 floats = 16 KB

  unsigned tid    = threadIdx.x;
  unsigned waveId = tid >> 5;
  unsigned lane   = tid & 31u;
  unsigned m      = lane & 15u;                // A-row / D-column (sample)
  unsigned half   = lane >> 4;                 // K pair select {0,1}/{2,3}

  unsigned wgRowBase = blockIdx.x * 128u;      // 128 rows per block (even)
  unsigned row = wgRowBase + waveId * 16u + m; // this lane's A row

  // ---- A matrix: two control points (K = 2*half, 2*half+1) ----
  unsigned b   = row / 126u;                   // once per lane, off hot path
  unsigned rem = row - b * 126u;
  unsigned seg = rem >> 1;
  unsigned c   = rem & 1u;
  unsigned i0  = seg + half * 2u;
  unsigned i1  = i0 + 1u;
  i0 = (i0 >= 63u) ? (i0 - 63u) : i0;          // (seg+k) % 63
  i1 = (i1 >= 63u) ? (i1 - 63u) : i1;
  v2f A;
  A.x = Q[(size_t)b * 128u + i0 * 2u + c];
  A.y = Q[(size_t)b * 128u + i1 * 2u + c];

  // ---- B matrices: cubic B-spline basis columns ----
  float p0, p1, p2, p3;
  basis4((float)m * (1.f / 19.f), p0, p1, p2, p3);              // samples 0..15
  v2f B0;
  B0.x = half ? p2 : p0;
  B0.y = half ? p3 : p1;

  float q0v, q1v, q2v, q3v;
  basis4((float)(16u + m) * (1.f / 19.f), q0v, q1v, q2v, q3v);  // samples 16..19
  bool valid = (m < 4u);                       // columns 4..15 forced to zero
  v2f B1;
  B1.x = valid ? (half ? q2v : q0v) : 0.f;
  B1.y = valid ? (half ? q3v : q1v) : 0.f;

  // ---- D = A x B (C = 0) ----
  v8f Cz = {};
  v8f D0 = __builtin_amdgcn_wmma_f32_16x16x4_f32(
      false, A, false, B0, (short)0, Cz, false, false);
  v8f D1 = __builtin_amdgcn_wmma_f32_16x16x4_f32(
      false, A, false, B1, (short)0, Cz, false, false);

  // ---- stage to LDS in (padded) output order ----
  // local pair index for store j: rows (waveId*16 + 8*half + 2j, +1)
  unsigned pBase = waveId * 8u + half * 4u;
#pragma unroll
  for (int j = 0; j < 4; ++j) {
    unsigned p = pBase + (unsigned)j;
    float2 v0 = make_float2(D0[2 * j], D0[2 * j + 1]);   // c=0,1 adjacent
    *(float2*)&lds[p * 64u + 2u * m] = v0;               // samples 0..15
    float2 v1 = make_float2(D1[2 * j], D1[2 * j + 1]);
    *(float2*)&lds[p * 64u + 32u + 2u * m] = v1;         // samples 16..31(pad)
  }
  __syncthreads();

  // ---- coalesced copy-out: 64 pairs x 40 valid floats = 640 float4 ----
  size_t outBase = (size_t)wgRowBase * 20u;
  for (unsigned idx = tid; idx < 640u; idx += 256u) {
    unsigned p = idx / 10u;
    unsigned q = idx - p * 10u;
    float4 v = *(const float4*)&lds[p * 64u + q * 4u];
    *(float4*)&out[outBase + (size_t)p * 40u + (size_t)q * 4u] = v;
  }
}

extern "C" void kernel_launch(void* const* d_in, const int* in_sizes, int n_in,
                              void* d_out, int out_size, void* d_ws,
                              size_t ws_size, hipStream_t stream) {
  const float* in = (const float*)d_in[0];
  float* out = (float*)d_out;
  float* Q = (float*)d_ws;              // [B,64,2] f32 = 8 MB scratch

  int B  = in_sizes[0] / 128;           // 16384 (N=64, C=2)
  int BC = B * 2;
  solve_ctrl_pts<<<(BC + 255) / 256, 256, 0, stream>>>(in, Q, BC);

  int nRows   = B * 126;                // B * 63 segs * 2 channels
  int nBlocks = nRows / 128;            // 128 rows per block (exact: 16128)
  bspline_wmma<<<nBlocks, 256, 0, stream>>>(Q, out);
}